// AELoss_1597727834374
// MI455X (gfx1250) — compile-verified
//
#include <hip/hip_runtime.h>
#include <hip/hip_bf16.h>

#define AE_B 32
#define AE_N (17 * 256 * 256)
#define AE_P 30
#define AE_J 17

typedef __attribute__((ext_vector_type(2))) float v2f;
typedef __attribute__((ext_vector_type(8))) float v8f;

__device__ __forceinline__ float wave_sum32(float v) {
#pragma unroll
    for (int off = 16; off > 0; off >>= 1)
        v += __shfl_xor(v, off, 32);
    return v;
}

// One wave32 block per batch element. lane == person index (lanes 30,31 masked
// arithmetically, never via EXEC, so WMMA sees EXEC == all ones).
__global__ void __launch_bounds__(32)
ae_loss_kernel(const float* __restrict__ tags,
               const int*   __restrict__ joints,
               float* __restrict__ out) {
    const int b    = blockIdx.x;
    const int lane = threadIdx.x;          // 0..31
    const bool act = (lane < AE_P);
    const int  p   = act ? lane : 0;       // clamp so loads stay in-range

    // ---- gather per-joint tags, accumulate mean ---------------------------
    const int2* jt = (const int2*)joints;  // (flat_idx, vis) pairs
    const long  jbase = (long)(b * AE_P + p) * AE_J;
    const float* tb = tags + (long)b * AE_N;

    float g[AE_J];
    float vf[AE_J];
    float sum = 0.f, cnt = 0.f;
#pragma unroll
    for (int j = 0; j < AE_J; ++j) {
        int2 e = jt[jbase + j];                    // independent b64 loads
        float v = (act && e.y > 0) ? 1.f : 0.f;
        float t = tb[e.x];                         // gather (always in-range)
        g[j]  = t;
        vf[j] = v;
        sum += t * v;
        cnt += v;
    }
    const float safe   = fmaxf(cnt, 1.f);
    const float mean   = sum / safe;               // == 0 when cnt == 0
    const float validf = (cnt > 0.f) ? 1.f : 0.f;

    // ---- pull -------------------------------------------------------------
    float pp = 0.f;
#pragma unroll
    for (int j = 0; j < AE_J; ++j) {
        float d = g[j] - mean;
        pp += d * d * vf[j];
    }
    pp = (pp / safe) * validf;

    const float pull_sum = wave_sum32(pp);
    const float num_tags = wave_sum32(validf);
    const float pull     = pull_sum / fmaxf(num_tags, 1.f);

    // ---- broadcast means/masks for the pairwise stage ---------------------
    __shared__ float s_mean[32];
    __shared__ float s_mask[32];
    s_mean[lane] = mean;    // lanes 30,31 hold 0 / 0
    s_mask[lane] = validf;
    __syncthreads();

    const int lo16  = lane & 15;
    const int hiSel = (lane >> 4) << 3;            // +8 rows for lanes 16..31

    // Unconditional LDS preloads (all indices in-range) -> fragment builds
    // become pure v_cndmask selects, no EXEC-predicated loads before WMMA.
    const float mh[2] = { s_mean[lo16], s_mean[16 + lo16] };   // col means
    const float kh[2] = { s_mask[lo16], s_mask[16 + lo16] };   // col masks
    float mrow[2][8], krow[2][8];                               // row means/masks
#pragma unroll
    for (int r = 0; r < 8; ++r) {
        mrow[0][r] = s_mean[hiSel + r];
        krow[0][r] = s_mask[hiSel + r];
        mrow[1][r] = s_mean[16 + hiSel + r];
        krow[1][r] = s_mask[16 + hiSel + r];
    }
    const bool loHalf = (lane < 16);

    // ---- push: 32x32 outer product mean_i*mean_j via V_WMMA_F32_16X16X4_F32
    // exp(-(mi-mj)^2) = exp(2*mi*mj - mi^2 - mj^2); mi*mj from WMMA tiles.
    float push_acc = 0.f;
#pragma unroll
    for (int ti = 0; ti < 2; ++ti) {
#pragma unroll
        for (int tj = 0; tj < 2; ++tj) {
            // A: 16x4, only K=0 column non-zero (lanes 0-15 carry K=0/1)
            v2f a;  a.x  = loHalf ? mh[ti] : 0.f;  a.y  = 0.f;
            // B: 4x16, only K=0 row non-zero (VGPR0, lanes 0-15 carry K=0)
            v2f bf; bf.x = loHalf ? mh[tj] : 0.f;  bf.y = 0.f;
            v8f c = {};
            c = __builtin_amdgcn_wmma_f32_16x16x4_f32(
                    false, a, false, bf, (short)0, c, false, false);

            const float mc  = mh[tj];
            const float kc  = kh[tj];
            const float mc2 = mc * mc;
#pragma unroll
            for (int r = 0; r < 8; ++r) {
                const float mr = mrow[ti][r];
                const float kr = krow[ti][r];
                push_acc += kr * kc * __expf(2.f * c[r] - mr * mr - mc2);
            }
        }
    }

    const float push_total = wave_sum32(push_acc);   // includes diagonal (=num_tags)

    if (lane == 0) {
        const float push_sum = push_total - num_tags;
        const float denom    = fmaxf((num_tags - 1.f) * num_tags, 1.f);
        const float push     = (num_tags > 1.f) ? (push_sum / denom * 0.5f) : 0.f;
        out[b]        = push;   // tuple order: (push, pull)
        out[AE_B + b] = pull;
    }
}

extern "C" void kernel_launch(void* const* d_in, const int* in_sizes, int n_in,
                              void* d_out, int out_size, void* d_ws, size_t ws_size,
                              hipStream_t stream) {
    const float* tags   = (const float*)d_in[0];
    const int*   joints = (const int*)d_in[1];
    float*       out    = (float*)d_out;
    (void)in_sizes; (void)n_in; (void)out_size; (void)d_ws; (void)ws_size;

    ae_loss_kernel<<<dim3(AE_B), dim3(32), 0, stream>>>(tags, joints, out);
}